// HexPatchLayer_9577777070774
// MI455X (gfx1250) — compile-verified
//
#include <hip/hip_runtime.h>
#include <math.h>

// HexPatchLayer for MI455X (gfx1250, wave32).
// Memory-bound gather workload: image set (128MB) is L2-resident (192MB L2),
// output 25MB. One 256-thread block per (batch b, sub-lattice k); each thread
// keeps its 64 patch samples in VGPRs, wave-reduces sum/sumsq via
// V_WMMA_F32_16X16X4_F32 (all-ones B => row sums), finishes across 8 waves in
// LDS, then normalizes and stores with a non-temporal hint.

#define PP 128
#define SQ3_2 0.8660254037844386f

typedef float v2f __attribute__((ext_vector_type(2)));
typedef float v8f __attribute__((ext_vector_type(8)));

// Full-wave (32 lane) sum using the WMMA pipe.
// A: lane L holds A[?,k0]=v (VGPR0), A[?,k1]=0 (VGPR1). B = all ones.
// D[m,n] = val(lane m) + val(lane m+16) for every n. Summing the 8 D VGPRs in
// a lane covers rows M=0..7 (lanes<16) or M=8..15 (lanes>=16); one xor-16
// shuffle merges the two halves -> total sum broadcast to every lane.
__device__ __forceinline__ float wave_sum_wmma(float v) {
    v2f a; a[0] = v;    a[1] = 0.0f;
    v2f bm; bm[0] = 1.0f; bm[1] = 1.0f;
    v8f c = {0.0f, 0.0f, 0.0f, 0.0f, 0.0f, 0.0f, 0.0f, 0.0f};
    v8f d = __builtin_amdgcn_wmma_f32_16x16x4_f32(
        /*neg_a=*/false, a, /*neg_b=*/false, bm,
        /*c_mod=*/(short)0, c, /*reuse_a=*/false, /*reuse_b=*/false);
    float s = d[0] + d[1] + d[2] + d[3] + d[4] + d[5] + d[6] + d[7];
    s += __shfl_xor(s, 16, 32);
    return s;
}

__global__ __launch_bounds__(256)
void hexpatch_kernel(const float* __restrict__ image,
                     const float* __restrict__ scale,
                     const float* __restrict__ hexshape,
                     const float* __restrict__ pos,
                     const float* __restrict__ hexrot,
                     float* __restrict__ out,
                     int H, int W) {
    const int k = blockIdx.x;   // hex sub-lattice 0..2
    const int b = blockIdx.y;   // batch
    const int t = threadIdx.x;  // 0..255

    // Per-image scalars (uniform across the block -> scalar loads).
    const float sc  = fminf(fmaxf(scale[b], 1.0f), 4.0f);
    const float hs  = hexshape[b];
    const float px  = fminf(fmaxf(pos[2 * b + 0], 0.0f), 1.0f);
    const float py  = fminf(fmaxf(pos[2 * b + 1], 0.0f), 1.0f);
    const float rot = hexrot[b];
    const float cr  = cosf(rot);
    const float sr  = sinf(rot);
    const float ext = sc * (float)PP;
    const float cx  = px * (float)(W - 1);
    const float cy  = py * (float)(H - 1);
    const float koff = (float)k / (3.0f * (float)PP);
    const float dlin = 1.0f / (float)(PP - 1);

    const float* __restrict__ img = image + (size_t)b * (size_t)H * (size_t)W;

    // Thread t owns column x = t % 128; rows y = 2*i + (t/128), i = 0..63.
    const int x = t & (PP - 1);
    const int yoff = t >> 7;
    const float u = -0.5f + (float)x * dlin + koff;   // loop-invariant

    float vals[64];
    float s = 0.0f, q = 0.0f;

#pragma unroll
    for (int i = 0; i < 64; ++i) {
        const int y = 2 * i + yoff;
        const float v  = -0.5f + (float)y * dlin + koff;
        const float hx = u + 0.5f * v;
        const float hy = SQ3_2 * v * hs;
        float gx = cx + (cr * hx - sr * hy) * ext;
        float gy = cy + (sr * hx + cr * hy) * ext;
        gx = fminf(fmaxf(gx, 0.0f), (float)(W - 1));
        gy = fminf(fmaxf(gy, 0.0f), (float)(H - 1));
        const float x0 = floorf(gx);
        const float y0 = floorf(gy);
        int x0i = (int)x0;
        int y0i = (int)y0;
        int x1i = x0i + 1; if (x1i > W - 1) x1i = W - 1;
        int y1i = y0i + 1; if (y1i > H - 1) y1i = H - 1;
        const float wx = gx - x0;
        const float wy = gy - y0;
        const float* __restrict__ r0 = img + (size_t)y0i * (size_t)W;
        const float* __restrict__ r1 = img + (size_t)y1i * (size_t)W;
        const float Ia = r0[x0i];
        const float Ib = r0[x1i];
        const float Ic = r1[x0i];
        const float Id = r1[x1i];
        const float top = Ia + wx * (Ib - Ia);
        const float bot = Ic + wx * (Id - Ic);
        const float val = top + wy * (bot - top);
        vals[i] = val;
        s += val;
        q += val * val;
    }

    // Wave-level reduction on the matrix pipe (EXEC all-1s here: no divergence).
    s = wave_sum_wmma(s);
    q = wave_sum_wmma(q);

    // Cross-wave (8 waves) reduction through LDS.
    __shared__ float redS[8];
    __shared__ float redQ[8];
    const int w = t >> 5;
    if ((t & 31) == 0) { redS[w] = s; redQ[w] = q; }
    __syncthreads();
    float S = 0.0f, Q = 0.0f;
#pragma unroll
    for (int j = 0; j < 8; ++j) { S += redS[j]; Q += redQ[j]; }

    const float n    = (float)(PP * PP);          // 16384
    const float mean = S / n;
    const float var  = fmaxf(Q / n - mean * mean, 0.0f);
    const float adj  = fmaxf(sqrtf(var), 1.0f / sqrtf(n));  // max(std, 1/128)
    const float inv  = 1.0f / adj;

    // out[b][y][x][k], layout [B, P, P, 3]; write-once -> non-temporal stores.
    float* __restrict__ ob = out + (size_t)b * PP * PP * 3 + k;
#pragma unroll
    for (int i = 0; i < 64; ++i) {
        const int y = 2 * i + yoff;
        const size_t idx = ((size_t)y * PP + (size_t)x) * 3;
        __builtin_nontemporal_store((vals[i] - mean) * inv, &ob[idx]);
    }
}

extern "C" void kernel_launch(void* const* d_in, const int* in_sizes, int n_in,
                              void* d_out, int out_size, void* d_ws, size_t ws_size,
                              hipStream_t stream) {
    (void)n_in; (void)d_ws; (void)ws_size; (void)out_size;
    const float* image    = (const float*)d_in[0];  // [B,H,W,1]
    const float* scale    = (const float*)d_in[1];  // [B,1]
    const float* hexshape = (const float*)d_in[2];  // [B,1]
    const float* pos      = (const float*)d_in[3];  // [B,2]
    const float* hexrot   = (const float*)d_in[4];  // [B,1]
    float* out = (float*)d_out;                     // [B,P,P,3]

    const int B  = in_sizes[1];
    const int HW = in_sizes[0] / B;
    const int H  = (int)lround(sqrt((double)HW));   // 512
    const int W  = H;

    dim3 grid(3, (unsigned)B);
    dim3 block(256);
    hexpatch_kernel<<<grid, block, 0, stream>>>(image, scale, hexshape, pos,
                                                hexrot, out, H, W);
}